// ChebyshevSheafFilter_71640054497900
// MI455X (gfx1250) — compile-verified
//
#include <hip/hip_runtime.h>
#include <hip/hip_bf16.h>

typedef float v2f __attribute__((ext_vector_type(2)));
typedef float v8f __attribute__((ext_vector_type(8)));

// ---------------------------------------------------------------------------
// Zero a float region (degree + off-diagonal scatter buffer).
// ---------------------------------------------------------------------------
__global__ __launch_bounds__(256) void sheaf_zero(float* __restrict__ p, int n) {
    int i = blockIdx.x * blockDim.x + threadIdx.x;
    if (i < n) p[i] = 0.0f;
}

// ---------------------------------------------------------------------------
// degree[n] = #incidences of n in row + col (matches the two index_adds).
// ---------------------------------------------------------------------------
__global__ __launch_bounds__(256) void sheaf_degree(const long long* __restrict__ rows,
                                                    const long long* __restrict__ cols,
                                                    float* __restrict__ deg, int E) {
    int e = blockIdx.x * blockDim.x + threadIdx.x;
    if (e < E) {
        atomicAdd(deg + rows[e], 1.0f);
        atomicAdd(deg + cols[e], 1.0f);
    }
}

// ---------------------------------------------------------------------------
// Per-edge 16x16 matvec via V_WMMA_F32_16X16X4_F32, scatter-add to off[].
// One wave32 per edge (grid-stride). A = Q[e][:,k:k+4], B = x[col] chunk
// broadcast to all 16 columns, so every column of the 16x16 accumulator holds
// y = Q[e] @ x[col[e]]. Q is streamed non-temporally (1 GB/pass, no reuse);
// x gathers stay cacheable (6.4 MB, L2-resident).
//
// A layout (32-bit A 16x4): lane m<16 holds {K0,K1}; lane m>=16 holds {K2,K3}
//   -> one contiguous b64 load per lane per K-chunk.
// B layout (4x16): v0 = row K0 (lanes 0-15) / K2 (lanes 16-31), v1 = K1/K3
//   -> same b64 pattern with the row index dropped (address shared in-half).
// D layout: lane n<16 holds y[0..7] in acc[0..7]; lanes>=16 hold y[8..15].
// ---------------------------------------------------------------------------
__global__ __launch_bounds__(256) void sheaf_edge_matvec(const float* __restrict__ Q,
                                                         const float* __restrict__ x,
                                                         const long long* __restrict__ rows,
                                                         const long long* __restrict__ cols,
                                                         float* __restrict__ off, int E) {
    const int lane  = threadIdx.x & 31;
    const int wave  = (int)((blockIdx.x * (blockDim.x >> 5)) + (threadIdx.x >> 5));
    const int nWave = (int)(gridDim.x * (blockDim.x >> 5));

    const int m     = lane & 15;          // A-matrix row this lane supplies
    const int khalf = (lane >> 4) << 1;   // 0 for lanes 0-15, 2 for lanes 16-31

    for (int e = wave; e < E; e += nWave) {
        const long long row = rows[e];
        const long long col = cols[e];
        const float* __restrict__ qe = Q + (long long)e * 256;
        const float* __restrict__ xe = x + col * 16;

        v8f acc = {0.f, 0.f, 0.f, 0.f, 0.f, 0.f, 0.f, 0.f};
#pragma unroll
        for (int c = 0; c < 4; ++c) {
            const int k = 4 * c + khalf;
            // Q slice: streamed once, keep out of caches (TH=NT).
            v2f a = __builtin_nontemporal_load((const v2f*)(qe + m * 16 + k));
            // x chunk broadcast: lanes 0-15 share one 8B line, 16-31 another.
            v2f b = *(const v2f*)(xe + k);
            acc = __builtin_amdgcn_wmma_f32_16x16x4_f32(
                false, a, false, b, (short)0, acc, false, false);
        }

        // Extract y: lanes 0-7 take acc[lane] (= y[lane]); lanes 16-23 take
        // acc[lane-16] (= y[8 + lane-16]). Unrolled select, no scratch.
        const int sel = lane & 7;
        float val = acc[0];
#pragma unroll
        for (int i = 1; i < 8; ++i) val = (sel == i) ? acc[i] : val;
        if ((lane & 15) < 8) {
            const int elem = (lane & 7) + ((lane >> 4) << 3);
            atomicAdd(off + row * 16 + elem, val);
        }
    }
}

// ---------------------------------------------------------------------------
// Elementwise Chebyshev step over [N*16]:
//   L   = scale*(deg[n]*x - off) - x,  scale = 2/(lambda_max+1e-8)
//   T_k = isFirst ? L : 2L - T_{k-2}
//   result (+)= softmax(coeffs)[k] * T_k   (pass 1 also adds w0*h, h == x)
// Re-zeros off[] in-place for the next pass (same thread read->write).
// ---------------------------------------------------------------------------
__global__ __launch_bounds__(256) void sheaf_combine(const float* __restrict__ x,
                                                     const float* __restrict__ prevprev,
                                                     float* __restrict__ off,
                                                     const float* __restrict__ deg,
                                                     const float* __restrict__ coeffs,
                                                     const float* __restrict__ lambda_max,
                                                     float* __restrict__ result,
                                                     float* __restrict__ tnext,
                                                     int n_elems, int widx, int isFirst) {
    int idx = blockIdx.x * blockDim.x + threadIdx.x;
    if (idx >= n_elems) return;

    // softmax over the 5 coefficients (tiny, L2/constant-cached reads)
    float c0 = coeffs[0], c1 = coeffs[1], c2 = coeffs[2], c3 = coeffs[3], c4 = coeffs[4];
    float mx = fmaxf(fmaxf(fmaxf(c0, c1), fmaxf(c2, c3)), c4);
    float e0 = __expf(c0 - mx), e1 = __expf(c1 - mx), e2 = __expf(c2 - mx),
          e3 = __expf(c3 - mx), e4 = __expf(c4 - mx);
    float inv = 1.0f / (e0 + e1 + e2 + e3 + e4);
    float w0 = e0 * inv, w1 = e1 * inv, w2 = e2 * inv, w3 = e3 * inv, w4 = e4 * inv;
    float wk = w1;
    wk = (widx == 2) ? w2 : wk;
    wk = (widx == 3) ? w3 : wk;
    wk = (widx == 4) ? w4 : wk;

    float scale = 2.0f / (lambda_max[0] + 1e-8f);
    int   node  = idx >> 4;
    float xv    = x[idx];
    float L     = scale * (deg[node] * xv - off[idx]) - xv;
    off[idx] = 0.0f;  // ready for next matvec pass

    float T;
    if (isFirst) {
        T = L;
        result[idx] = w0 * xv + wk * T;   // x == h on the first pass
    } else {
        T = 2.0f * L - prevprev[idx];
        result[idx] += wk * T;
    }
    tnext[idx] = T;
}

// ---------------------------------------------------------------------------
// Host-side orchestration (graph-capture safe: launches only).
// Inputs: h[N*16] f32, Q[E*256] f32, coeffs[5] f32, edge_index[2*E] i64,
//         lambda_max[1] f32.  Output: [N*16] f32.
// Workspace layout (floats): deg[N] | off[16N] | bufA[16N] | bufB[16N] | bufC[16N]
//   = 65N floats ~= 26 MB.
// ---------------------------------------------------------------------------
extern "C" void kernel_launch(void* const* d_in, const int* in_sizes, int n_in,
                              void* d_out, int out_size, void* d_ws, size_t ws_size,
                              hipStream_t stream) {
    const float*     h          = (const float*)d_in[0];
    const float*     Q          = (const float*)d_in[1];
    const float*     coeffs     = (const float*)d_in[2];
    const long long* eidx       = (const long long*)d_in[3];
    const float*     lambda_max = (const float*)d_in[4];

    const int N  = in_sizes[0] / 16;
    const int E  = in_sizes[3] / 2;
    const int NE = N * 16;

    const long long* rows = eidx;        // edge_index[0]
    const long long* cols = eidx + E;    // edge_index[1]

    float* ws   = (float*)d_ws;
    float* deg  = ws;
    float* off  = deg + N;
    float* bufA = off + NE;
    float* bufB = bufA + NE;
    float* bufC = bufB + NE;
    float* out  = (float*)d_out;

    const int TB = 256;

    // zero deg + off (adjacent 17N floats)
    {
        int zn = N + NE;
        sheaf_zero<<<(zn + TB - 1) / TB, TB, 0, stream>>>(deg, zn);
    }
    // degrees
    sheaf_degree<<<(E + TB - 1) / TB, TB, 0, stream>>>(rows, cols, deg, E);

    const int mvBlocks = 4096;  // 8 waves/block, grid-stride over 1M edges

    // Pass 1: T1 = L(h); result = w0*h + w1*T1
    sheaf_edge_matvec<<<mvBlocks, TB, 0, stream>>>(Q, h, rows, cols, off, E);
    sheaf_combine<<<(NE + TB - 1) / TB, TB, 0, stream>>>(h, nullptr, off, deg, coeffs,
                                                         lambda_max, out, bufA, NE, 1, 1);
    // Pass 2: T2 = 2L(T1) - h; result += w2*T2
    sheaf_edge_matvec<<<mvBlocks, TB, 0, stream>>>(Q, bufA, rows, cols, off, E);
    sheaf_combine<<<(NE + TB - 1) / TB, TB, 0, stream>>>(bufA, h, off, deg, coeffs,
                                                         lambda_max, out, bufB, NE, 2, 0);
    // Pass 3: T3 = 2L(T2) - T1; result += w3*T3
    sheaf_edge_matvec<<<mvBlocks, TB, 0, stream>>>(Q, bufB, rows, cols, off, E);
    sheaf_combine<<<(NE + TB - 1) / TB, TB, 0, stream>>>(bufB, bufA, off, deg, coeffs,
                                                         lambda_max, out, bufC, NE, 3, 0);
    // Pass 4: T4 = 2L(T3) - T2; result += w4*T4
    sheaf_edge_matvec<<<mvBlocks, TB, 0, stream>>>(Q, bufC, rows, cols, off, E);
    sheaf_combine<<<(NE + TB - 1) / TB, TB, 0, stream>>>(bufC, bufB, off, deg, coeffs,
                                                         lambda_max, out, bufA, NE, 4, 0);
}